// GINE_layer_53807350284451
// MI455X (gfx1250) — compile-verified
//
#include <hip/hip_runtime.h>

#define N_NODES  100000
#define N_EDGES  6400000
#define N_GRAPHS 1000

typedef float v2f __attribute__((ext_vector_type(2)));
typedef float v8f __attribute__((ext_vector_type(8)));

__device__ __forceinline__ void atomic_add_f32(float* p, float v) {
    // hardware global_atomic_add_f32 (no return)
    unsafeAtomicAdd(p, v);
}

// ---------------------------------------------------------------- fill zeros
__global__ void fill0_kernel(float* __restrict__ p, int n) {
    int i = blockIdx.x * blockDim.x + threadIdx.x;
    if (i < n) p[i] = 0.0f;
}

// ------------------------------------------------- edge message + scatter-add
// msg = relu(x[src] + edge_attr[e]); aggr[dst] += msg
// 4 threads per edge, each handles a float4 slice (fully coalesced edge_attr).
__global__ void edge_scatter_kernel(const float* __restrict__ xin,
                                    const int*   __restrict__ ei,   // [2, E]
                                    const float* __restrict__ ea,   // [E, 16]
                                    float*       __restrict__ aggr) // [N, 16]
{
    int idx = blockIdx.x * blockDim.x + threadIdx.x;
    if (idx >= N_EDGES * 4) return;
    int e = idx >> 2;
    int q = idx & 3;
    int s = ei[e];
    int d = ei[N_EDGES + e];

    const float4 xv = *reinterpret_cast<const float4*>(xin + (size_t)s * 16 + q * 4);
    const float4 ev = *reinterpret_cast<const float4*>(ea  + (size_t)e * 16 + q * 4);

    float4 m;
    m.x = fmaxf(xv.x + ev.x, 0.0f);
    m.y = fmaxf(xv.y + ev.y, 0.0f);
    m.z = fmaxf(xv.z + ev.z, 0.0f);
    m.w = fmaxf(xv.w + ev.w, 0.0f);

    float* ap = aggr + (size_t)d * 16 + q * 4;
    atomic_add_f32(ap + 0, m.x);
    atomic_add_f32(ap + 1, m.y);
    atomic_add_f32(ap + 2, m.z);
    atomic_add_f32(ap + 3, m.w);
}

// ----------------------------------------------------- 16x16x16 fp32 via WMMA
// One 16x16 GEMM tile per wave using 4x V_WMMA_F32_16X16X4_F32 (K=16).
// A layout (16x4 f32): lanes 0-15 -> M=lane, K=4k+{0,1}; lanes 16-31 -> K=4k+{2,3}
// B layout (4x16 f32): lanes 0-15 -> N=lane, K=4k+{0,1}; lanes 16-31 -> K=4k+{2,3}
// C/D layout: VGPR r -> row r (lanes 0-15) / row r+8 (lanes 16-31), col = lane&15
__device__ __forceinline__ v8f wmma_gemm16(const float* __restrict__ A,  // 16x16 row-major (LDS)
                                           const float* __restrict__ B,  // 16x16 row-major (LDS)
                                           int lane)
{
    v8f acc = {};
    const int m   = lane & 15;
    const int hi2 = (lane >> 4) << 1;   // 0 for lanes 0-15, 2 for lanes 16-31
#pragma unroll
    for (int k = 0; k < 4; ++k) {
        const int kb = 4 * k + hi2;
        v2f a, b;
        a.x = A[m * 16 + kb];
        a.y = A[m * 16 + kb + 1];
        b.x = B[kb * 16 + m];
        b.y = B[(kb + 1) * 16 + m];
        acc = __builtin_amdgcn_wmma_f32_16x16x4_f32(false, a, false, b,
                                                    (short)0, acc, false, false);
    }
    return acc;
}

// -------------------------------------------------------------- node MLP pass
// out = [relu?]( relu((x+aggr) @ Wa + ba) @ Wb + bb ), weights zero-padded to
// 16x16 in LDS so layer 2's 16x6 / 6x6 weights use the same WMMA path exactly.
__global__ __launch_bounds__(256)
void mlp_wmma_kernel(const float* __restrict__ xin,   // [N,16]
                     const float* __restrict__ aggr,  // [N,16]
                     const float* __restrict__ Wa, const float* __restrict__ ba, int ka_cols,
                     const float* __restrict__ Wb, const float* __restrict__ bb, int kb_in, int kb_cols,
                     float* __restrict__ out, int out_cols, int relu_out)
{
    __shared__ float sWa[256];
    __shared__ float sWb[256];
    __shared__ float sBa[16];
    __shared__ float sBb[16];
    __shared__ float sT[8][256];   // per-wave 16x16 activation tile

    const int tid  = threadIdx.x;
    const int wr   = tid >> 4, wc = tid & 15;
    sWa[tid] = (wc < ka_cols) ? Wa[wr * ka_cols + wc] : 0.0f;
    sWb[tid] = (wr < kb_in && wc < kb_cols) ? Wb[wr * kb_cols + wc] : 0.0f;
    if (tid < 16) {
        sBa[tid] = (tid < ka_cols) ? ba[tid] : 0.0f;
        sBb[tid] = (tid < kb_cols) ? bb[tid] : 0.0f;
    }
    __syncthreads();

    const int ntiles = N_NODES / 16;            // 6250, exact
    const int wave = tid >> 5;
    const int lane = tid & 31;
    const int tile = blockIdx.x * 8 + wave;
    const int tc   = tile < ntiles ? tile : ntiles - 1;   // clamp, no early return
    const size_t base = (size_t)tc * 256;
    float* T = sT[wave];

    // stage H = x + aggr (contiguous 256 floats, 8 per lane)
    {
        const int f = lane * 8;
        float4 a0 = *reinterpret_cast<const float4*>(xin  + base + f);
        float4 a1 = *reinterpret_cast<const float4*>(xin  + base + f + 4);
        float4 g0 = *reinterpret_cast<const float4*>(aggr + base + f);
        float4 g1 = *reinterpret_cast<const float4*>(aggr + base + f + 4);
        T[f + 0] = a0.x + g0.x;  T[f + 1] = a0.y + g0.y;
        T[f + 2] = a0.z + g0.z;  T[f + 3] = a0.w + g0.w;
        T[f + 4] = a1.x + g1.x;  T[f + 5] = a1.y + g1.y;
        T[f + 6] = a1.z + g1.z;  T[f + 7] = a1.w + g1.w;
    }
    // wave-internal LDS ops are in-order; no barrier needed for per-wave tile

    const int m     = lane & 15;
    const int mrow0 = (lane >> 4) * 8;

    // GEMM 1: H @ Wa, + bias, ReLU, back into the tile
    v8f acc = wmma_gemm16(T, sWa, lane);
    {
        const float bias = sBa[m];
#pragma unroll
        for (int rr = 0; rr < 8; ++rr) {
            float v = fmaxf(acc[rr] + bias, 0.0f);
            T[(mrow0 + rr) * 16 + m] = v;
        }
    }

    // GEMM 2: Hmid @ Wb, + bias, optional ReLU, store
    v8f acc2 = wmma_gemm16(T, sWb, lane);
    {
        const float bias = sBb[m];
        if (tile < ntiles && m < out_cols) {
#pragma unroll
            for (int rr = 0; rr < 8; ++rr) {
                float v = acc2[rr] + bias;
                if (relu_out) v = fmaxf(v, 0.0f);
                out[((size_t)tc * 16 + mrow0 + rr) * out_cols + m] = v;
            }
        }
    }
}

// ------------------------------------------------------------- graph pooling
__global__ void pool_kernel(const float* __restrict__ nout,   // [N,6]
                            const int*   __restrict__ batch,  // [N]
                            float* __restrict__ gsum,         // [G,6]
                            float* __restrict__ gcnt)         // [G]
{
    int i = blockIdx.x * blockDim.x + threadIdx.x;
    if (i >= N_NODES) return;
    int g = batch[i];
    const float* p = nout + (size_t)i * 6;
#pragma unroll
    for (int j = 0; j < 6; ++j) atomic_add_f32(&gsum[g * 6 + j], p[j]);
    atomic_add_f32(&gcnt[g], 1.0f);
}

// ------------------------------------------------- mean + log_softmax (G=1000)
__global__ void logsoftmax_kernel(const float* __restrict__ gsum,
                                  const float* __restrict__ gcnt,
                                  float* __restrict__ out)
{
    int g = blockIdx.x * blockDim.x + threadIdx.x;
    if (g >= N_GRAPHS) return;
    float cnt = fmaxf(gcnt[g], 1.0f);
    float v[6];
    float mx = -3.4e38f;
#pragma unroll
    for (int j = 0; j < 6; ++j) { v[j] = gsum[g * 6 + j] / cnt; mx = fmaxf(mx, v[j]); }
    float s = 0.0f;
#pragma unroll
    for (int j = 0; j < 6; ++j) s += expf(v[j] - mx);
    float lse = mx + logf(s);
#pragma unroll
    for (int j = 0; j < 6; ++j) out[g * 6 + j] = v[j] - lse;
}

// ---------------------------------------------------------------------- entry
extern "C" void kernel_launch(void* const* d_in, const int* in_sizes, int n_in,
                              void* d_out, int out_size, void* d_ws, size_t ws_size,
                              hipStream_t stream) {
    const float* x     = (const float*)d_in[0];
    const int*   ei    = (const int*)  d_in[1];
    const float* ea    = (const float*)d_in[2];
    const int*   batch = (const int*)  d_in[3];
    const float* W1a   = (const float*)d_in[4];
    const float* b1a   = (const float*)d_in[5];
    const float* W1b   = (const float*)d_in[6];
    const float* b1b   = (const float*)d_in[7];
    const float* W2a   = (const float*)d_in[8];
    const float* b2a   = (const float*)d_in[9];
    const float* W2b   = (const float*)d_in[10];
    const float* b2b   = (const float*)d_in[11];
    float* out = (float*)d_out;

    float* ws   = (float*)d_ws;
    float* aggr = ws;                                 // N*16
    float* h    = aggr + (size_t)N_NODES * 16;        // N*16
    float* nout = h    + (size_t)N_NODES * 16;        // N*6
    float* gsum = nout + (size_t)N_NODES * 6;         // G*6
    float* gcnt = gsum + (size_t)N_GRAPHS * 6;        // G   (contiguous with gsum)

    const int B = 256;
    const int nAggr   = N_NODES * 16;
    const int gFill   = (nAggr + B - 1) / B;
    const int gEdge   = (N_EDGES * 4 + B - 1) / B;
    const int gMlp    = (N_NODES / 16 + 7) / 8;       // 8 waves (tiles) per block
    const int gPool   = (N_NODES + B - 1) / B;
    const int gFinal  = (N_GRAPHS + B - 1) / B;
    const int gFillG  = (N_GRAPHS * 7 + B - 1) / B;

    // ---- layer 1
    fill0_kernel<<<gFill, B, 0, stream>>>(aggr, nAggr);
    edge_scatter_kernel<<<gEdge, B, 0, stream>>>(x, ei, ea, aggr);
    mlp_wmma_kernel<<<gMlp, B, 0, stream>>>(x, aggr, W1a, b1a, 16,
                                            W1b, b1b, 16, 16, h, 16, /*relu_out=*/1);
    // ---- layer 2
    fill0_kernel<<<gFill, B, 0, stream>>>(aggr, nAggr);
    edge_scatter_kernel<<<gEdge, B, 0, stream>>>(h, ei, ea, aggr);
    mlp_wmma_kernel<<<gMlp, B, 0, stream>>>(h, aggr, W2a, b2a, 6,
                                            W2b, b2b, 6, 6, nout, 6, /*relu_out=*/0);
    // ---- pooling + log_softmax
    fill0_kernel<<<gFillG, B, 0, stream>>>(gsum, N_GRAPHS * 7);  // gsum + gcnt
    pool_kernel<<<gPool, B, 0, stream>>>(nout, batch, gsum, gcnt);
    logsoftmax_kernel<<<gFinal, B, 0, stream>>>(gsum, gcnt, out);
}